// LogScale_35287451304310
// MI455X (gfx1250) — compile-verified
//
#include <hip/hip_runtime.h>
#include <hip/hip_bf16.h>
#include <math.h>

#define N_IN 2049

typedef unsigned int v4u __attribute__((ext_vector_type(4)));
typedef int          v8i __attribute__((ext_vector_type(8)));
typedef int          v4i __attribute__((ext_vector_type(4)));

// ---------------------------------------------------------------------------
// Pass 1: per triangular-output row, find the contiguous window of finite
// weights (everything else is -inf). Runs once per launch; ~289 blocks.
// bounds[2t] = first finite index, bounds[2t+1] = last finite index + 1.
// ---------------------------------------------------------------------------
__global__ __launch_bounds__(256) void tri_bounds_kernel(
    const float* __restrict__ w, int* __restrict__ bounds) {
  const int t = blockIdx.x;
  const float* wr = w + (size_t)t * N_IN;
  int lo = N_IN, hi = -1;
  for (int k = threadIdx.x; k < N_IN; k += 256) {
    float v = wr[k];
    if (v >= -1.0e30f) {          // finite (weights are in [-12, 0] dB)
      if (k < lo) lo = k;
      if (k > hi) hi = k;
    }
  }
  __shared__ int slo[256];
  __shared__ int shi[256];
  slo[threadIdx.x] = lo;
  shi[threadIdx.x] = hi;
  __syncthreads();
  for (int s = 128; s > 0; s >>= 1) {
    if ((int)threadIdx.x < s) {
      slo[threadIdx.x] = min(slo[threadIdx.x], slo[threadIdx.x + s]);
      shi[threadIdx.x] = max(shi[threadIdx.x], shi[threadIdx.x + s]);
    }
    __syncthreads();
  }
  if (threadIdx.x == 0) {
    bounds[2 * t]     = slo[0];
    bounds[2 * t + 1] = shi[0] + 1;
  }
}

// ---------------------------------------------------------------------------
// Pass 2: one workgroup per input row. TDM-DMA the 2049-float row into LDS,
// then each thread produces output bins (linear lerp / cubic / windowed
// max-plus) reading exclusively from LDS.
// ---------------------------------------------------------------------------
__global__ __launch_bounds__(256) void logscale_kernel(
    const float* __restrict__ x,
    const float* __restrict__ flin,
    const float* __restrict__ fcub,
    const float* __restrict__ w,
    const int*   __restrict__ pidx,
    const int*   __restrict__ bounds,
    float*       __restrict__ out,
    int n_lin, int n_cub, int n_tri) {
  __shared__ float srow[N_IN];   // single shared array: 8196 B of 320 KB/WGP

  const int row   = blockIdx.x;
  const int n_out = n_lin + n_cub + n_tri;

  // ---- async tensor DMA: global row -> LDS (issued by wave 0 only) ----
  unsigned long long gaddr =
      (unsigned long long)(const void*)(x + (size_t)row * N_IN);
  unsigned lds_off = (unsigned)(size_t)(void*)srow;  // low 32 bits of flat
                                                     // LDS addr = LDS offset
  int wid = __builtin_amdgcn_readfirstlane((int)(threadIdx.x >> 5));
  if (wid == 0) {  // uniform scalar branch: waves 1..7 skip (TDM ignores EXEC)
    v4u g0;
    g0[0] = 1u;                                       // count=1, user D#
    g0[1] = lds_off;                                  // lds_addr[31:0]
    g0[2] = (unsigned)(gaddr & 0xFFFFFFFFull);        // global_addr[31:0]
    g0[3] = (unsigned)((gaddr >> 32) & 0x1FFFFFFull)  // global_addr[56:32]
            | (2u << 30);                             // type = 2 (image)
    v8i g1;
    g1[0] = (int)(2u << 16);                 // data_size = 2 -> 4 bytes
    g1[1] = (int)((unsigned)N_IN << 16);     // tensor_dim0[15:0] @ bits 63:48
    g1[2] = (int)(1u << 16);                 // tensor_dim1[15:0] @ bits 95:80
    g1[3] = (int)((unsigned)N_IN << 16);     // tile_dim0         @ bits 127:112
    g1[4] = 1;                               // tile_dim1         @ bits 143:128
    g1[5] = N_IN;                            // tensor_dim0_stride[31:0]
    g1[6] = 0;
    g1[7] = 0;
    v4i gz4  = {0, 0, 0, 0};                 // <=2D tensor: groups 2/3 unused
    v8i gz8  = {0, 0, 0, 0, 0, 0, 0, 0};     // 6-arg form: extra group, zeroed
    __builtin_amdgcn_tensor_load_to_lds(g0, g1, gz4, gz4, gz8, 0);
    __builtin_amdgcn_s_wait_tensorcnt(0);
  }
  __syncthreads();   // publish LDS row to all 8 waves

  // ---- compute: 1024 outputs per row, all gathers hit LDS ----
  for (int j = threadIdx.x; j < n_out; j += 256) {
    float r;
    if (j < n_lin) {
      // linear: x0 + f*(x1-x0)
      int   i0 = pidx[j];
      int   i1 = pidx[n_lin + j];
      float f  = flin[j];
      float a  = srow[i0];
      float b  = srow[i1];
      r = a + f * (b - a);
    } else if (j < n_lin + n_cub) {
      // Catmull-Rom style cubic, same association order as reference
      float fc  = fcub[j - n_lin];
      int   i0  = (int)floorf(fc);
      float f   = fc - (float)i0;
      float xm1 = srow[i0 - 1];
      float x0  = srow[i0];
      float x1  = srow[i0 + 1];
      float x2  = srow[i0 + 2];
      r = x0 + 0.5f * f *
              (x1 - xm1 +
               f * (2.0f * xm1 - 5.0f * x0 + 4.0f * x1 - x2 +
                    f * (3.0f * (x0 - x1) + x2 - xm1)));
    } else {
      // windowed max-plus: max_k (x[k] + w[t,k]) over the finite window
      int t = j - n_lin - n_cub;
      int s = bounds[2 * t];
      int e = bounds[2 * t + 1];
      const float* wr = w + (size_t)t * N_IN;
      float m = -INFINITY;
      for (int k = s; k < e; ++k)
        m = fmaxf(m, srow[k] + wr[k]);
      r = m;
    }
    out[(size_t)row * n_out + j] = r;
  }
}

extern "C" void kernel_launch(void* const* d_in, const int* in_sizes, int n_in_args,
                              void* d_out, int out_size, void* d_ws, size_t ws_size,
                              hipStream_t stream) {
  const float* x    = (const float*)d_in[0];
  const float* flin = (const float*)d_in[1];
  const float* fcub = (const float*)d_in[2];
  const float* w    = (const float*)d_in[3];
  const int*   pidx = (const int*)d_in[4];

  const int n_lin = in_sizes[1];
  const int n_cub = in_sizes[2];
  const int n_tri = in_sizes[3] / N_IN;
  const int rows  = in_sizes[0] / N_IN;      // 32*512 = 16384

  int* bounds = (int*)d_ws;                  // 2*n_tri ints of scratch

  if (n_tri > 0)
    tri_bounds_kernel<<<n_tri, 256, 0, stream>>>(w, bounds);

  logscale_kernel<<<rows, 256, 0, stream>>>(
      x, flin, fcub, w, pidx, bounds, (float*)d_out, n_lin, n_cub, n_tri);
}